// MyModel_79147657331347
// MI455X (gfx1250) — compile-verified
//
#include <hip/hip_runtime.h>

#define N_AG   1024
#define T_IN   50
#define TOUT   20
#define FMAXX  128
#define IN_F   8
#define OUT_F  32
#define HID    64
#define GATES  256
#define DTT    0.04f
#define AGB    16      // agents per block (one WMMA M-tile)
#define THREADS 128    // 4 wave32
#define CHUNK  10      // LSTM timesteps per X-precompute chunk

typedef __attribute__((ext_vector_type(2))) float v2f;
typedef __attribute__((ext_vector_type(8))) float v8f;

__device__ __forceinline__ v8f wmma4(v2f a, v2f b, v8f c) {
    // V_WMMA_F32_16X16X4_F32 : exact fp32 D = A(16x4) * B(4x16) + C
    return __builtin_amdgcn_wmma_f32_16x16x4_f32(
        /*neg_a=*/false, a, /*neg_b=*/false, b,
        /*c_mod=*/(short)0, c, /*reuse_a=*/false, /*reuse_b=*/false);
}

// Branch-free activations: v_exp_f32 (exp2) + v_rcp_f32, no exec-mask branching.
__device__ __forceinline__ float fast_sigm(float x) {
    return __builtin_amdgcn_rcpf(1.0f + __builtin_amdgcn_exp2f(-1.44269504f * x));
}
__device__ __forceinline__ float fast_tanh(float x) {
    return 2.0f * __builtin_amdgcn_rcpf(1.0f + __builtin_amdgcn_exp2f(-2.88539008f * x)) - 1.0f;
}

// -------- kernel 0: zero the frame tables in workspace --------
__global__ void zero_ws_kernel(float* ws, int n) {
    int i = blockIdx.x * blockDim.x + threadIdx.x;
    if (i < n) ws[i] = 0.0f;
}

// -------- kernel 1: global frame tables (segment_sum) --------
__global__ void frame_tables_kernel(const float* __restrict__ inp,
                                    float* __restrict__ cnt,
                                    float* __restrict__ sums) {
    int i = blockIdx.x * blockDim.x + threadIdx.x;
    if (i < N_AG * T_IN) {
        const float* row = inp + (size_t)i * 6;
        int f = (int)row[0];
        atomicAdd(&cnt[f], 1.0f);
        atomicAdd(&sums[f * 4 + 0], row[2]);
        atomicAdd(&sums[f * 4 + 1], row[3]);
        atomicAdd(&sums[f * 4 + 2], row[4]);
        atomicAdd(&sums[f * 4 + 3], row[5]);
    }
}

// -------- kernel 2: main rollout, 16 agents per workgroup --------
__global__ __launch_bounds__(THREADS)
void traj_rollout_kernel(const float* __restrict__ inp,   // (N,T,6)
                         const float* __restrict__ tgt,   // (N,TOUT,6)
                         const float* __restrict__ W1,    // (32,8)
                         const float* __restrict__ b1,    // (32)
                         const float* __restrict__ W2,    // (32,32)
                         const float* __restrict__ b2,    // (32)
                         const float* __restrict__ Wih,   // (256,32)
                         const float* __restrict__ Whh,   // (256,64)
                         const float* __restrict__ bih,   // (256)
                         const float* __restrict__ bhh,   // (256)
                         const float* __restrict__ scale, // (4)
                         const float* __restrict__ gCnt,  // (128)
                         const float* __restrict__ gSums, // (128,4)
                         float* __restrict__ out)         // (N,TOUT,4)
{
    __shared__ float sSeq[AGB][T_IN][IN_F];       // 25.6 KB ring-buffered pooled seq
    __shared__ float sWhh[GATES][HID];            // 64 KB recurrent weights
    __shared__ float sH[2][AGB][HID];             // 8 KB  ping-pong hidden state
    __shared__ float sXc[AGB][CHUNK][OUT_F];      // 20 KB chunk of LSTM inputs
    __shared__ float sHid2[2][AGB][OUT_F];        // 4 KB  MLP layer-1 staging
    __shared__ float sCnt[FMAXX];
    __shared__ float sSums[FMAXX][4];
    __shared__ float sScale[4];

    const int tid  = threadIdx.x;
    const int lane = tid & 31;
    const int wv   = tid >> 5;       // wave id 0..3 (uniform per wave)
    const int hl   = lane >> 4;      // half-wave select (K-pair / M-block)
    const int nn   = lane & 15;      // N index within tile
    const int mm   = lane & 15;      // M (agent) index for A-fragments

    // warm the weight lines (global_prefetch_b8)
    __builtin_prefetch(Whh, 0, 1);
    __builtin_prefetch(Wih, 0, 1);
    __builtin_prefetch(W2, 0, 1);

    // ---- stage recurrent weights + tables into LDS ----
    for (int i = tid; i < GATES * HID; i += THREADS) ((float*)sWhh)[i] = Whh[i];
    for (int i = tid; i < FMAXX; i += THREADS) sCnt[i] = gCnt[i];
    for (int i = tid; i < FMAXX * 4; i += THREADS) ((float*)sSums)[i] = gSums[i];
    if (tid < 4) sScale[tid] = scale[tid];
    __syncthreads();

    // ---- initial pooled sequence for this block's 16 agents ----
    const int ag0 = blockIdx.x * AGB;
    for (int i = tid; i < AGB * T_IN; i += THREADS) {
        int a = i / T_IN, t = i % T_IN;
        const float* row = inp + ((size_t)(ag0 + a) * T_IN + t) * 6;
        int f = (int)row[0];
        #pragma unroll
        for (int q = 0; q < 4; q++) {
            float v = row[2 + q];
            sSeq[a][t][q]     = v;
            sSeq[a][t][4 + q] = v * sCnt[f] - sSums[f][q];
        }
    }

    // ---- hoist loop-invariant B-fragments / biases / scales into registers ----
    const int mrow  = wv >> 1;       // row slot within MLP pass (0/1)
    const int mn0   = 16 * (wv & 1); // MLP N-tile base
    v2f rW1[2], rW2[8];
    #pragma unroll
    for (int kk = 0; kk < 2; kk++) {
        int k = 4 * kk + 2 * hl, n = mn0 + nn;
        rW1[kk] = v2f{ W1[n * IN_F + k], W1[n * IN_F + k + 1] };
    }
    #pragma unroll
    for (int kk = 0; kk < 8; kk++) {
        int k = 4 * kk + 2 * hl, n = mn0 + nn;
        rW2[kk] = v2f{ W2[n * OUT_F + k], W2[n * OUT_F + k + 1] };
    }
    const float rb1 = b1[mn0 + nn];
    const float rb2 = b2[mn0 + nn];
    v2f rWih[4][8];
    float rbg[4];
    #pragma unroll
    for (int gg = 0; gg < 4; gg++) {
        int n = 64 * gg + 16 * wv + nn;
        rbg[gg] = bih[n] + bhh[n];
        #pragma unroll
        for (int kk = 0; kk < 8; kk++) {
            int k = 4 * kk + 2 * hl;
            rWih[gg][kk] = v2f{ Wih[n * OUT_F + k], Wih[n * OUT_F + k + 1] };
        }
    }
    const float sc0 = scale[0], sc1 = scale[1], sc2 = scale[2], sc3 = scale[3];
    __syncthreads();

    int base = 0; // ring-buffer origin
    for (int step = 0; step < TOUT; step++) {
        // ---- reset LSTM state: c in registers, h buffer 0 in LDS ----
        v8f cfrag = {0.f, 0.f, 0.f, 0.f, 0.f, 0.f, 0.f, 0.f};
        int pp = 0;
        for (int i = tid; i < AGB * HID; i += THREADS) ((float*)sH)[i] = 0.0f;
        __syncthreads();

        #pragma unroll 1
        for (int ch = 0; ch < T_IN / CHUNK; ch++) {
            const int rbase = ch * CHUNK;

            // ---- MLP: precompute X for CHUNK rows, all 4 waves busy ----
            #pragma unroll 1
            for (int pass = 0; pass < CHUNK / 2; pass++) {
                const int ri = pass * 2 + mrow;
                int p = base + rbase + ri; if (p >= T_IN) p -= T_IN;
                { // layer 1: tanh(seq @ W1.T + b1)
                    v8f acc = {rb1, rb1, rb1, rb1, rb1, rb1, rb1, rb1};
                    #pragma unroll
                    for (int kk = 0; kk < 2; kk++) {
                        const int kb = 4 * kk + 2 * hl;
                        v2f a = v2f{ sSeq[mm][p][kb], sSeq[mm][p][kb + 1] };
                        acc = wmma4(a, rW1[kk], acc);
                    }
                    #pragma unroll
                    for (int j = 0; j < 8; j++)
                        sHid2[mrow][j + 8 * hl][mn0 + nn] = fast_tanh(acc[j]);
                }
                __syncthreads();
                { // layer 2: X = hid @ W2.T + b2
                    v8f acc = {rb2, rb2, rb2, rb2, rb2, rb2, rb2, rb2};
                    #pragma unroll
                    for (int kk = 0; kk < 8; kk++) {
                        const int kb = 4 * kk + 2 * hl;
                        v2f a = v2f{ sHid2[mrow][mm][kb], sHid2[mrow][mm][kb + 1] };
                        acc = wmma4(a, rW2[kk], acc);
                    }
                    #pragma unroll
                    for (int j = 0; j < 8; j++)
                        sXc[j + 8 * hl][ri][mn0 + nn] = acc[j];
                }
                __syncthreads();
            }

            // ---- LSTM: CHUNK sequential cell steps, 1 barrier each ----
            #pragma unroll 1
            for (int tt = 0; tt < CHUNK; tt++) {
                // 4 independent accumulate chains (i,f,g,o) interleaved:
                // each A-fragment is loaded once and feeds all 4 gates.
                v8f g[4];
                #pragma unroll
                for (int gg = 0; gg < 4; gg++)
                    g[gg] = v8f{rbg[gg], rbg[gg], rbg[gg], rbg[gg],
                                rbg[gg], rbg[gg], rbg[gg], rbg[gg]};
                #pragma unroll
                for (int kk = 0; kk < 16; kk++) {      // h @ Whh.T
                    const int kb = 4 * kk + 2 * hl;
                    v2f a = v2f{ sH[pp][mm][kb], sH[pp][mm][kb + 1] };
                    #pragma unroll
                    for (int gg = 0; gg < 4; gg++) {
                        const int n = 64 * gg + 16 * wv + nn;
                        v2f b = v2f{ sWhh[n][kb], sWhh[n][kb + 1] };
                        g[gg] = wmma4(a, b, g[gg]);
                    }
                }
                #pragma unroll
                for (int kk = 0; kk < 8; kk++) {       // x @ Wih.T (reg B-frags)
                    const int kb = 4 * kk + 2 * hl;
                    v2f a = v2f{ sXc[mm][tt][kb], sXc[mm][tt][kb + 1] };
                    #pragma unroll
                    for (int gg = 0; gg < 4; gg++)
                        g[gg] = wmma4(a, rWih[gg][kk], g[gg]);
                }
                // cell update, elementwise, branch-free; write to other h buffer
                #pragma unroll
                for (int j = 0; j < 8; j++) {
                    float iv = fast_sigm(g[0][j]);
                    float fv = fast_sigm(g[1][j]);
                    float gv = fast_tanh(g[2][j]);
                    float ov = fast_sigm(g[3][j]);
                    float cv = fv * cfrag[j] + iv * gv;
                    cfrag[j] = cv;
                    sH[pp ^ 1][j + 8 * hl][16 * wv + nn] = ov * fast_tanh(cv);
                }
                __syncthreads();
                pp ^= 1;
            }
        }

        // ---- scale first 4 feature columns of all rows (contiguous quads) ----
        for (int i = tid; i < AGB * T_IN; i += THREADS) {
            int a = i / T_IN, r = i % T_IN;
            float* rp = &sSeq[a][r][0];
            rp[0] *= sc0; rp[1] *= sc1; rp[2] *= sc2; rp[3] *= sc3;
        }
        __syncthreads();

        // ---- physics + output + append pooled row (1 thread per agent) ----
        if (tid < AGB) {
            const int a = tid;
            const int gA = ag0 + a;
            int pl = base + T_IN - 1; if (pl >= T_IN) pl -= T_IN;
            float x  = sSeq[a][pl][0], y  = sSeq[a][pl][1];
            float vx = sSeq[a][pl][2], vy = sSeq[a][pl][3];
            float a0 = sH[pp][a][0], a1 = sH[pp][a][1];
            float vxp = vx + DTT * a0;
            float vyp = vy + DTT * a1;
            float xp  = x + DTT * vxp;
            float yp  = y + DTT * vyp;
            float frame = tgt[((size_t)gA * TOUT + step) * 6 + 0];
            float fx  = xp  / sc0;
            float fy  = yp  / sc1;
            float fvx = vxp / sc2;
            float fvy = vyp / sc3;
            float* o = out + ((size_t)gA * TOUT + step) * 4;
            o[0] = fx; o[1] = fy; o[2] = fvx; o[3] = fvy;
            int fr = (int)frame;
            float cN = sCnt[fr];
            float* dst = &sSeq[a][base][0];   // evicted row becomes new last row
            dst[0] = fx;  dst[1] = fy;  dst[2] = fvx; dst[3] = fvy;
            dst[4] = fx  * cN - sSums[fr][0];
            dst[5] = fy  * cN - sSums[fr][1];
            dst[6] = fvx * cN - sSums[fr][2];
            dst[7] = fvy * cN - sSums[fr][3];
        }
        __syncthreads();
        base++; if (base >= T_IN) base = 0;
    }
}

extern "C" void kernel_launch(void* const* d_in, const int* in_sizes, int n_in,
                              void* d_out, int out_size, void* d_ws, size_t ws_size,
                              hipStream_t stream) {
    const float* inp   = (const float*)d_in[0];
    const float* tgt   = (const float*)d_in[1];
    const float* W1    = (const float*)d_in[2];
    const float* b1    = (const float*)d_in[3];
    const float* W2    = (const float*)d_in[4];
    const float* b2    = (const float*)d_in[5];
    const float* Wih   = (const float*)d_in[6];
    const float* Whh   = (const float*)d_in[7];
    const float* bih   = (const float*)d_in[8];
    const float* bhh   = (const float*)d_in[9];
    const float* scale = (const float*)d_in[10];
    float* out = (float*)d_out;

    float* ws   = (float*)d_ws;
    float* cnt  = ws;               // 128 floats
    float* sums = ws + FMAXX;       // 512 floats

    const int tab = FMAXX * 5;
    zero_ws_kernel<<<(tab + 255) / 256, 256, 0, stream>>>(ws, tab);
    frame_tables_kernel<<<(N_AG * T_IN + 255) / 256, 256, 0, stream>>>(inp, cnt, sums);
    traj_rollout_kernel<<<N_AG / AGB, THREADS, 0, stream>>>(
        inp, tgt, W1, b1, W2, b2, Wih, Whh, bih, bhh, scale, cnt, sums, out);
}